// SPDTransform_13546326851712
// MI455X (gfx1250) — compile-verified
//
#include <hip/hip_runtime.h>

// out[b] = W^T @ X[b] @ W
// B=2048, N=256, M=128, all f32.
// One workgroup per batch: 16 waves (512 threads, wave32).
// Stage 1: T = X[b] @ W   (256x128) -> LDS
// Stage 2: out = W^T @ T  (128x128) -> global
// WMMA: V_WMMA_F32_16X16X4_F32 (f32 matrix pipe, 2048 FLOP/instr).

#define BATCH 2048
#define NN 256
#define MM 128

typedef __attribute__((ext_vector_type(2))) float v2f;
typedef __attribute__((ext_vector_type(8))) float v8f;

__global__ __launch_bounds__(512) void spd_congruence_kernel(
    const float* __restrict__ X,   // [B, 256, 256]
    const float* __restrict__ W,   // [256, 128]
    float* __restrict__ out)       // [B, 128, 128]
{
    // k-pair interleaved layouts: s[kp][m] = (A[2kp][m], A[2kp+1][m])
    __shared__ v2f sW[128 * 128];  // 128 KB
    __shared__ v2f sT[128 * 128];  // 128 KB

    const int b      = blockIdx.x;
    const int tid    = threadIdx.x;
    const int wave   = tid >> 5;
    const int lane   = tid & 31;
    const int laneLo = lane & 15;   // row/col within 16
    const int laneHi = lane >> 4;   // 0: K pair (k,k+1); 1: (k+2,k+3)

    // ---- stage W into interleaved LDS (coalesced b32 loads, b64 LDS stores)
    for (int idx = tid; idx < 128 * 128; idx += 512) {
        const int kp = idx >> 7;     // k-pair index 0..127
        const int m  = idx & 127;
        v2f wv;
        wv.x = W[(2 * kp + 0) * MM + m];
        wv.y = W[(2 * kp + 1) * MM + m];
        sW[idx] = wv;
    }
    __syncthreads();

    // ---- stage 1: T = X[b] @ W ----
    const float* Xb = X + (size_t)b * NN * NN;
    const int r0 = wave * 16;                // each wave: rows r0..r0+15 of T

    v8f acc[8] = {};                          // 8 n-tiles of 16x16 f32
    for (int k = 0; k < NN; k += 4) {
        const int kp = (k >> 1) + laneHi;     // lane's k-pair row
        // A fragment: X rows r0+laneLo, cols (k..k+1 | k+2..k+3)
        v2f a = *(const v2f*)(Xb + (r0 + laneLo) * NN + k + 2 * laneHi);
#pragma unroll
        for (int j = 0; j < 8; ++j) {
            v2f bb = sW[kp * 128 + j * 16 + laneLo];
            acc[j] = __builtin_amdgcn_wmma_f32_16x16x4_f32(
                false, a, false, bb, (short)0, acc[j], false, false);
        }
    }

    // D layout: VGPR v holds row (r0 + v + 8*laneHi), col (n0 + laneLo).
    // Row pairs (2t, 2t+1) live in the same lane -> b64 stores into sT.
#pragma unroll
    for (int j = 0; j < 8; ++j) {
        const int n = j * 16 + laneLo;
#pragma unroll
        for (int t = 0; t < 4; ++t) {
            const int kp = (r0 >> 1) + 4 * laneHi + t;
            v2f tv;
            tv.x = acc[j][2 * t + 0];
            tv.y = acc[j][2 * t + 1];
            sT[kp * 128 + n] = tv;
        }
    }
    __syncthreads();

    // ---- stage 2: out[b] = W^T @ T ----
    // 64 tiles (8x8) split 4 per wave.
    const int m0 = (wave & 7) * 16;
    const int nb = (wave >> 3) * 64;

    v8f acc2[4] = {};
    for (int k = 0; k < NN; k += 4) {
        const int kp = (k >> 1) + laneHi;
        // A fragment: Wt[m0+laneLo][k..] == W[k..][m0+laneLo] -> same sW layout
        v2f a = sW[kp * 128 + m0 + laneLo];
#pragma unroll
        for (int j = 0; j < 4; ++j) {
            v2f bb = sT[kp * 128 + nb + j * 16 + laneLo];
            acc2[j] = __builtin_amdgcn_wmma_f32_16x16x4_f32(
                false, a, false, bb, (short)0, acc2[j], false, false);
        }
    }

    float* Ob = out + (size_t)b * MM * MM;
#pragma unroll
    for (int j = 0; j < 4; ++j) {
        const int col = nb + j * 16 + laneLo;
#pragma unroll
        for (int v = 0; v < 8; ++v) {
            const int row = m0 + v + 8 * laneHi;
            Ob[row * MM + col] = acc2[j][v];   // coalesced across lanes 0..15
        }
    }
}

extern "C" void kernel_launch(void* const* d_in, const int* in_sizes, int n_in,
                              void* d_out, int out_size, void* d_ws, size_t ws_size,
                              hipStream_t stream) {
    const float* X = (const float*)d_in[0];   // [B,256,256] f32
    const float* W = (const float*)d_in[1];   // [256,128]  f32
    float* out = (float*)d_out;               // [B,128,128] f32

    const int B = in_sizes[0] / (NN * NN);    // 2048
    spd_congruence_kernel<<<B, 512, 0, stream>>>(X, W, out);
}